// CTCLayer_22325240004776
// MI455X (gfx1250) — compile-verified
//
#include <hip/hip_runtime.h>
#include <cstdint>

// Problem constants (from reference setup_inputs)
#define CTC_B   512
#define CTC_T   512
#define CTC_C   128
#define CTC_L   64
#define CTC_S   129          // 2L+1
#define CTC_BLANK 127        // C-1
#define NEGINF  (-1e30f)
#define CEPS    (1e-7f)

#define DEPTH    16          // async ring depth per wave (power of 2)
#define NTHREADS 160         // 5 wave32s; states 0..128 live on threads 0..128
#define NWAVES   (NTHREADS / 32)

__global__ __launch_bounds__(NTHREADS, 1)
void ctc_forward_kernel(const int* __restrict__ y_true,
                        const float* __restrict__ y_pred,
                        float* __restrict__ loss)
{
    // Per-wave private row ring: 5 waves x 16 slots x 128 floats = 40 KB
    __shared__ __align__(16) float rowbuf[NWAVES * DEPTH * CTC_C];
    // Double-buffered alpha with 2-entry NEG pad at the front (state s at index s+2)
    __shared__ float alpha[2][CTC_S + 3];

    const int b    = blockIdx.x;
    const int tid  = threadIdx.x;
    const int wave = tid >> 5;
    const int lane = tid & 31;
    const int s    = tid;

    const float* gbase = y_pred + (size_t)b * (CTC_T * CTC_C);

    // LDS byte address of this wave's ring (lane handles 16B of each 512B row)
    const float*  wavebuf  = &rowbuf[wave * DEPTH * CTC_C];
    const uint32_t lds_wave = (uint32_t)(uintptr_t)wavebuf + (uint32_t)(lane * 16);

    // Per-state constants: extended label + skip-transition mask
    int  cls  = CTC_BLANK;
    bool skip = false;
    if (s < CTC_S) {
        if (s & 1) {
            const int li = s >> 1;
            cls = y_true[b * CTC_L + li];
            if (li > 0) skip = (cls != y_true[b * CTC_L + li - 1]);
        }
    }

    // Front pads stay NEG forever
    if (tid < 2) { alpha[0][tid] = NEGINF; alpha[1][tid] = NEGINF; }

    // ---- Prologue: asynchronously stage rows 0..DEPTH-2 into this wave's ring ----
    #pragma unroll
    for (int r = 0; r < DEPTH - 1; ++r) {
        const uint32_t voff  = (uint32_t)(r * CTC_C * 4) + (uint32_t)(lane * 16);
        const uint32_t laddr = lds_wave + (uint32_t)(r * CTC_C * 4);
        asm volatile("global_load_async_to_lds_b128 %0, %1, %2"
                     :: "v"(laddr), "v"(voff), "s"(gbase) : "memory");
    }

    // ---- t = 0: issue row DEPTH-1, wait for row 0, initialize alpha ----
    {
        const uint32_t voff  = (uint32_t)((DEPTH - 1) * CTC_C * 4) + (uint32_t)(lane * 16);
        const uint32_t laddr = lds_wave + (uint32_t)((DEPTH - 1) * CTC_C * 4);
        asm volatile("global_load_async_to_lds_b128 %0, %1, %2"
                     :: "v"(laddr), "v"(voff), "s"(gbase) : "memory");
        asm volatile("s_wait_asynccnt 15" ::: "memory");   // 16 outstanding -> row 0 done
        if (s < CTC_S) {
            float a = NEGINF;
            if (s < 2) a = __logf(wavebuf[cls] + CEPS);    // slot 0
            alpha[0][s + 2] = a;
        }
    }
    __syncthreads();

    float* Acur = alpha[0];
    float* Anxt = alpha[1];

    // ---- Forward recurrence over time ----
    for (int t = 1; t < CTC_T; ++t) {
        // Issue prefetch for row t+DEPTH-1 (clamped dummy near the end keeps the
        // outstanding count at exactly DEPTH so the wait below always means
        // "row t has landed"). Target slot (t-1)%DEPTH was consumed last step
        // by this same wave only, so no cross-wave hazard.
        {
            const int pfr  = t + DEPTH - 1;
            const int slot = pfr & (DEPTH - 1);
            const int row  = (pfr < CTC_T) ? pfr : (CTC_T - 1);
            const uint32_t voff  = (uint32_t)(row * CTC_C * 4) + (uint32_t)(lane * 16);
            const uint32_t laddr = lds_wave + (uint32_t)(slot * CTC_C * 4);
            asm volatile("global_load_async_to_lds_b128 %0, %1, %2"
                         :: "v"(laddr), "v"(voff), "s"(gbase) : "memory");
        }
        asm volatile("s_wait_asynccnt 15" ::: "memory");   // row t is now in LDS

        if (s < CTC_S) {
            const float* rp = wavebuf + (t & (DEPTH - 1)) * CTC_C;
            const float lp = __logf(rp[cls] + CEPS);

            const float a0 = Acur[s + 2];                  // alpha[s]
            const float a1 = Acur[s + 1];                  // alpha[s-1]
            const float a2 = skip ? Acur[s] : NEGINF;      // alpha[s-2] if allowed
            const float m  = fmaxf(a0, fmaxf(a1, a2));
            const float sm = __expf(a0 - m) + __expf(a1 - m) + __expf(a2 - m);
            Anxt[s + 2] = m + __logf(sm) + lp;
        }
        __syncthreads();
        float* tmp = Acur; Acur = Anxt; Anxt = tmp;
    }

    // ---- Termination: end in last blank (S-1) or last label (S-2) ----
    if (tid == 0) {
        const float aN  = Acur[CTC_S + 1];   // state S-1
        const float aN1 = Acur[CTC_S];       // state S-2
        const float m   = fmaxf(aN, aN1);
        loss[b] = -(m + __logf(__expf(aN - m) + __expf(aN1 - m)));
    }
}

extern "C" void kernel_launch(void* const* d_in, const int* in_sizes, int n_in,
                              void* d_out, int out_size, void* d_ws, size_t ws_size,
                              hipStream_t stream) {
    (void)in_sizes; (void)n_in; (void)out_size; (void)d_ws; (void)ws_size;

    const int*   y_true = (const int*)d_in[0];    // [B, L] int32
    const float* y_pred = (const float*)d_in[1];  // [B, T, C] float32
    float*       out    = (float*)d_out;          // [B] loss, then [B,T,C] passthrough

    // Tuple output part 2: y_pred passthrough — pure bandwidth, let the copy
    // engine / driver path do it at full HBM rate (graph-capture safe).
    hipMemcpyAsync(out + CTC_B, y_pred,
                   (size_t)CTC_B * CTC_T * CTC_C * sizeof(float),
                   hipMemcpyDeviceToDevice, stream);

    // Tuple output part 1: per-example CTC NLL.
    ctc_forward_kernel<<<CTC_B, NTHREADS, 0, stream>>>(y_true, y_pred, out);
}